// PositiveLayer_48137993454056
// MI455X (gfx1250) — compile-verified
//
#include <hip/hip_runtime.h>

// C^4 for 2,000,000 batched 6x6 fp32 matrices via V_WMMA_F32_16X16X4_F32.
// Two matrices (P,Q) per 16x16 tile (block-diagonal: P rows/cols 0..5,
// Q rows/cols 8..13, K blocks 0..5 / 6..11); squaring = 3 WMMAs (K=12).
// Each wave processes 8 pairs; lane geometry, masks, LDS addresses AND the
// global base pointers of every address class are hoisted, so the loop body
// is loads/stores with immediate offsets + cndmask + wmma only.

typedef __attribute__((ext_vector_type(2))) float v2f;
typedef __attribute__((ext_vector_type(4))) float f4;
typedef __attribute__((ext_vector_type(8))) float v8f;

#define NTHREADS 256
#define WAVES_PER_BLOCK 8
#define PAIRS_PER_WAVE 8          // 8 waves * 8 pairs * 2 = 128 matrices / block
#define PAIR_BYTES 288            // 72 floats

__device__ __forceinline__ v2f msk(v2f v, bool cond) {
  v2f r; r.x = cond ? v.x : 0.0f; r.y = cond ? v.y : 0.0f; return r;
}

__device__ __forceinline__ v8f wmma3(v2f a0, v2f b0, v2f a1, v2f b1, v2f a2, v2f b2) {
  v8f acc = {0.0f, 0.0f, 0.0f, 0.0f, 0.0f, 0.0f, 0.0f, 0.0f};
  acc = __builtin_amdgcn_wmma_f32_16x16x4_f32(false, a0, false, b0, (short)0, acc, false, false);
  acc = __builtin_amdgcn_wmma_f32_16x16x4_f32(false, a1, false, b1, (short)0, acc, false, false);
  acc = __builtin_amdgcn_wmma_f32_16x16x4_f32(false, a2, false, b2, (short)0, acc, false, false);
  return acc;
}

__global__ __launch_bounds__(NTHREADS) void pow4_6x6_wmma(
    const float* __restrict__ C, float* __restrict__ out, int nmat) {
  // Per-wave scratch: A-panel 16x12 + B-panel 12x16 (zero-padded), 1536 B.
  __shared__ __align__(16) float panels[WAVES_PER_BLOCK * 384];

  const int tid  = threadIdx.x;
  const int lane = tid & 31;
  const int wave = tid >> 5;

  // ---- pair-invariant lane geometry ----
  const int mn   = lane & 15;
  const int half = lane >> 4;
  const int mp   = mn < 5 ? mn : 5;
  int tq = mn - 8; tq = tq < 0 ? 0 : tq;
  const int mq   = tq < 5 ? tq : 5;
  const bool vP  = mn < 6;
  const bool vQ  = (mn >= 8) & (mn < 14);
  const bool v1  = half ? vQ : vP;

  // ---- pair-invariant lane offsets (floats, relative to pair base) ----
  const int offA0 = mp * 6 + 2 * half;                       // slice0 A (b64)
  const int offA1 = half ? (36 + mq * 6) : (mp * 6 + 4);     // slice1 A (b64)
  const int offA2 = 36 + mq * 6 + 2 + 2 * half;              // slice2 A (b64)
  const int offB0 = (2 * half) * 6 + mp;                     // slice0 B (+0,+24B)
  const int offB1 = half ? (36 + mq) : (24 + mp);            // slice1 B (+0,+24B)
  const int offB2 = 36 + (2 + 2 * half) * 6 + mq;            // slice2 B (+0,+24B)
  const int offO  = half ? (mn + 28) : mn;                   // store base (+v*24B)

  // ---- pair-invariant LDS addresses ----
  float* pan  = panels + wave * 384;
  float* panA = pan;                                         // 16 x 12
  float* panB = pan + 192;                                   // 12 x 16
  const int kA = half ? (mn - 2) : mn;
  float* scA = panA + (8 * half) * 12 + kA;                  // scatter, +v*48B
  float* scB = panB + (6 * half) * 16 + mn;                  // scatter, +v*64B
  const float* rA = panA + mn * 12 + 2 * half;               // gather, +{0,16,32}B
  const float* rB = panB + (2 * half) * 16 + mn;             // gather

  // Zero panels once; scatter hits the same slots every pair, padding persists.
  { f4 zz = {0.0f, 0.0f, 0.0f, 0.0f};
    f4* z = (f4*)pan; z[lane] = zz; z[lane + 32] = zz; z[lane + 64] = zz; }
  asm volatile("" ::: "memory");

  const int waveIdx = __builtin_amdgcn_readfirstlane(
      (int)blockIdx.x * WAVES_PER_BLOCK + wave);
  const long long pair0 = (long long)waveIdx * PAIRS_PER_WAVE;
  if (pair0 * 2 >= (long long)nmat) return;                  // wave-uniform

  if ((pair0 + PAIRS_PER_WAVE) * 2 <= (long long)nmat) {
    // ================= fast path: 8 fully-populated pairs =================
    const float* wb = C + pair0 * 72;
    float* wo = out + pair0 * 72;
    // One 64-bit vaddr per address class; loop uses immediate offsets only.
    const char* gA0 = (const char*)(wb + offA0);
    const char* gA1 = (const char*)(wb + offA1);
    const char* gA2 = (const char*)(wb + offA2);
    const char* gB0 = (const char*)(wb + offB0);
    const char* gB1 = (const char*)(wb + offB1);
    const char* gB2 = (const char*)(wb + offB2);
    char*       gO  = (char*)(wo + offO);

#pragma unroll
    for (int it = 0; it < PAIRS_PER_WAVE; ++it) {
      const int ib = it * PAIR_BYTES;
      // stage 1: operands straight from global, cndmask-zeroed
      v2f a0 = msk(*(const v2f*)(gA0 + ib), vP);
      v2f b0; b0.x = *(const float*)(gB0 + ib); b0.y = *(const float*)(gB0 + ib + 24);
      b0 = msk(b0, vP);
      v2f a1 = msk(*(const v2f*)(gA1 + ib), v1);
      v2f b1; b1.x = *(const float*)(gB1 + ib); b1.y = *(const float*)(gB1 + ib + 24);
      b1 = msk(b1, v1);
      v2f a2 = msk(*(const v2f*)(gA2 + ib), vQ);
      v2f b2; b2.x = *(const float*)(gB2 + ib); b2.y = *(const float*)(gB2 + ib + 24);
      b2 = msk(b2, vQ);
      v8f c2 = wmma3(a0, b0, a1, b1, a2, b2);

      // reshuffle C^2 into padded panels (one exec-mask, 12 imm-offset stores)
      if (v1) {
#pragma unroll
        for (int v = 0; v < 6; ++v) { scA[v * 12] = c2[v]; scB[v * 16] = c2[v]; }
      }
      asm volatile("" ::: "memory");                         // scatter -> reads

      // stage 2: unconditional LDS reads (imm offsets)
      v2f A0 = *(const v2f*)(rA);
      v2f A1 = *(const v2f*)(rA + 4);
      v2f A2 = *(const v2f*)(rA + 8);
      v2f B0; B0.x = rB[0];   B0.y = rB[16];
      v2f B1; B1.x = rB[64];  B1.y = rB[80];
      v2f B2; B2.x = rB[128]; B2.y = rB[144];
      v8f c4 = wmma3(A0, B0, A1, B1, A2, B2);
      asm volatile("" ::: "memory");                         // reads -> next scatter

      if (v1) {
#pragma unroll
        for (int v = 0; v < 6; ++v) *(float*)(gO + ib + v * 24) = c4[v];
      }
    }
  } else {
    // ============== generic tail: per-pair, not unrolled, Q guard ==========
#pragma unroll 1
    for (int it = 0; it < PAIRS_PER_WAVE; ++it) {
      const long long pair = pair0 + it;
      if (pair * 2 >= (long long)nmat) break;                // wave-uniform
      const bool hasQ = (pair * 2 + 1) < (long long)nmat;
      const float* pBp = C + pair * 72;
      const float* qBp = hasQ ? (pBp + 36) : pBp;

      v2f a0 = msk(*(const v2f*)(pBp + mp * 6 + 2 * half), vP);
      v2f b0; b0.x = pBp[(2 * half) * 6 + mp]; b0.y = pBp[(2 * half + 1) * 6 + mp];
      b0 = msk(b0, vP);
      const float* a1p = half ? (qBp + mq * 6) : (pBp + mp * 6 + 4);
      v2f a1 = msk(*(const v2f*)a1p, v1);
      const float* b1p = half ? (qBp + mq) : (pBp + 24 + mp);
      v2f b1; b1.x = b1p[0]; b1.y = b1p[6]; b1 = msk(b1, v1);
      v2f a2 = msk(*(const v2f*)(qBp + mq * 6 + 2 + 2 * half), vQ);
      const float* b2p = qBp + (2 + 2 * half) * 6 + mq;
      v2f b2; b2.x = b2p[0]; b2.y = b2p[6]; b2 = msk(b2, vQ);
      v8f c2 = wmma3(a0, b0, a1, b1, a2, b2);

      if (v1) {
#pragma unroll
        for (int v = 0; v < 6; ++v) { scA[v * 12] = c2[v]; scB[v * 16] = c2[v]; }
      }
      asm volatile("" ::: "memory");

      v2f A0 = *(const v2f*)(rA);
      v2f A1 = *(const v2f*)(rA + 4);
      v2f A2 = *(const v2f*)(rA + 8);
      v2f B0; B0.x = rB[0];   B0.y = rB[16];
      v2f B1; B1.x = rB[64];  B1.y = rB[80];
      v2f B2; B2.x = rB[128]; B2.y = rB[144];
      v8f c4 = wmma3(A0, B0, A1, B1, A2, B2);
      asm volatile("" ::: "memory");

      float* os = out + pair * 72 + offO;
      const bool vs = v1 && (hasQ || !half);
      if (vs) {
#pragma unroll
        for (int v = 0; v < 6; ++v) os[v * 6] = c4[v];
      }
    }
  }
}

extern "C" void kernel_launch(void* const* d_in, const int* in_sizes, int n_in,
                              void* d_out, int out_size, void* d_ws, size_t ws_size,
                              hipStream_t stream) {
  (void)n_in; (void)d_ws; (void)ws_size; (void)out_size;
  const float* C = (const float*)d_in[0];
  float* out = (float*)d_out;
  const int nmat = in_sizes[0] / 36;                         // 2,000,000
  const long long npairs = ((long long)nmat + 1) / 2;        // 1,000,000
  const long long pairsPerBlock = (long long)WAVES_PER_BLOCK * PAIRS_PER_WAVE;
  const int blocks = (int)((npairs + pairsPerBlock - 1) / pairsPerBlock);  // 15,625
  pow4_6x6_wmma<<<blocks, NTHREADS, 0, stream>>>(C, out, nmat);
}